// DeformableCONV_31095563223341
// MI455X (gfx1250) — compile-verified
//
#include <hip/hip_runtime.h>
#include <stdint.h>

// Deformable depthwise conv, B=32 C=17 H=96 W=72 K=3 PAD=1.
// Memory-bound (~435 MB moved, AI~0.85) -> streaming kernel:
//  - x[b,c] plane staged into LDS via CDNA5 async global->LDS DMA (ASYNCcnt)
//  - offsets/mask streamed with 128-bit non-temporal loads (read-once data)
//  - bilinear gathers served from LDS (ds_load_b32), not scattered VMEM

#define BDIM 256

constexpr int Bn = 32, Cn = 17, Hn = 96, Wn = 72;
constexpr int Kn = 3, KKn = 9, PADn = 1;
constexpr int NPIX = Hn * Wn;   // 6912 pixels per (b,c) plane
constexpr int NV4  = NPIX / 4;  // 1728 float4 groups (W%4==0 -> no row crossing)

typedef float v4f __attribute__((ext_vector_type(4)));

__device__ __forceinline__ v4f nt_load4(const float* p) {
  return __builtin_nontemporal_load((const v4f*)p);
}
__device__ __forceinline__ void nt_store4(float* p, v4f v) {
  __builtin_nontemporal_store(v, (v4f*)p);
}

__global__ __launch_bounds__(BDIM) void deform_dwconv_kernel(
    const float* __restrict__ x,        // [B,C,H,W]
    const float* __restrict__ offsets,  // [B,2*C*KK,H,W]  (C,KK,2) channel order
    const float* __restrict__ mask,     // [B,C*KK,H,W]
    const float* __restrict__ weight,   // [C,1,K,K] -> [C,KK]
    const float* __restrict__ bias,     // [C]
    float* __restrict__ out)            // [B,C,H,W]
{
  __shared__ float tile[NPIX];  // 27,648 B of the 320 KB WGP LDS

  const int bc  = blockIdx.x;       // 0..B*C-1
  const int b   = bc / Cn;
  const int c   = bc - b * Cn;
  const int tid = threadIdx.x;

  // ---- Stage x[b,c,:,:] into LDS with async global->LDS b128 DMA ----
  {
    const float* gsrc = x + (size_t)bc * NPIX;
    for (int i = tid; i < NV4; i += BDIM) {
      uint32_t ldsa  = (uint32_t)(uintptr_t)(&tile[i * 4]); // low 32b of flat LDS addr
      const float* ga = gsrc + i * 4;
      asm volatile("global_load_async_to_lds_b128 %0, %1, off"
                   :: "v"(ldsa), "v"(ga)
                   : "memory");
    }
#if defined(__has_builtin) && __has_builtin(__builtin_amdgcn_s_wait_asynccnt)
    __builtin_amdgcn_s_wait_asynccnt(0);
#else
    asm volatile("s_wait_asynccnt 0" ::: "memory");
#endif
  }
  __syncthreads();

  // ---- Per-channel taps (block-uniform -> scalarized by compiler) ----
  float wk[KKn];
#pragma unroll
  for (int k = 0; k < KKn; ++k) wk[k] = weight[c * KKn + k];
  const float bs = bias[c];

  const float* offc = offsets + ((size_t)b * (2 * Cn * KKn)) * NPIX;
  const float* mc   = mask    + ((size_t)b * (Cn * KKn)) * NPIX;
  float*       outc = out     + (size_t)bc * NPIX;

  // Bilinear corner: contribution zero outside the image (matches reference);
  // read index clamped so the LDS access is always in range.
  auto corner = [&](int yi, int xi, float wgt) -> float {
    const bool valid = ((unsigned)yi < (unsigned)Hn) & ((unsigned)xi < (unsigned)Wn);
    const int yc = min(max(yi, 0), Hn - 1);
    const int xc = min(max(xi, 0), Wn - 1);
    return valid ? tile[yc * Wn + xc] * wgt : 0.0f;
  };

  // ---- Main stream: each thread owns 4 row-contiguous pixels per step ----
  for (int vp = tid; vp < NV4; vp += BDIM) {
    const int p0 = vp * 4;
    const int h  = p0 / Wn;
    const int w0 = p0 - h * Wn;   // group never crosses a row (72 % 4 == 0)

    v4f acc = {bs, bs, bs, bs};

#pragma unroll
    for (int k = 0; k < KKn; ++k) {
      const int ky = k / Kn;
      const int kx = k - ky * Kn;
      const int ch = c * KKn + k;

      // read-once streams: 128-bit non-temporal, wave-coalesced
      const v4f dy4 = nt_load4(offc + (size_t)(2 * ch + 0) * NPIX + p0);
      const v4f dx4 = nt_load4(offc + (size_t)(2 * ch + 1) * NPIX + p0);
      const v4f m4  = nt_load4(mc   + (size_t)ch * NPIX + p0);

#pragma unroll
      for (int j = 0; j < 4; ++j) {
        const float py  = (float)(h + ky - PADn) + dy4[j];
        const float px  = (float)(w0 + j + kx - PADn) + dx4[j];
        const float y0f = floorf(py);
        const float x0f = floorf(px);
        const float wy  = py - y0f;
        const float wx  = px - x0f;
        const int   y0  = (int)y0f;
        const int   x0  = (int)x0f;

        const float v = corner(y0,     x0,     (1.0f - wy) * (1.0f - wx))
                      + corner(y0,     x0 + 1, (1.0f - wy) * wx)
                      + corner(y0 + 1, x0,     wy * (1.0f - wx))
                      + corner(y0 + 1, x0 + 1, wy * wx);

        acc[j] = fmaf(v * m4[j], wk[k], acc[j]);
      }
    }
    nt_store4(outc + p0, acc);
  }
}

extern "C" void kernel_launch(void* const* d_in, const int* in_sizes, int n_in,
                              void* d_out, int out_size, void* d_ws, size_t ws_size,
                              hipStream_t stream) {
  const float* x       = (const float*)d_in[0];
  const float* offsets = (const float*)d_in[1];
  const float* mask    = (const float*)d_in[2];
  const float* weight  = (const float*)d_in[3];
  const float* bias    = (const float*)d_in[4];
  float* out = (float*)d_out;

  dim3 grid(Bn * Cn);   // 544 blocks of 8 wave32s
  deform_dwconv_kernel<<<grid, BDIM, 0, stream>>>(x, offsets, mask, weight, bias, out);

  (void)in_sizes; (void)n_in; (void)out_size; (void)d_ws; (void)ws_size;
}